// TurboEncoder_61727269978291
// MI455X (gfx1250) — compile-verified
//
#include <hip/hip_runtime.h>
#include <stdint.h>

// ---------------------------------------------------------------------------
// Turbo encoder (rate 1/3, g0=1+D^2, g1=1+D+D^2) for MI455X / gfx1250.
//
// Key identity: with this polynomial pair, parity p_t = u_t XOR
// (XOR of u_j for j < t with j of opposite parity to t).  So the whole
// encoder is two parity-class prefix-XOR streams -> fully parallel:
//   * per-lane word-level prefix via 5 shift-XOR steps per 32 bits
//   * cross-lane exclusive prefix via wave32 ballot+popcount
// Memory-bound: ~250MB traffic @ 23.3 TB/s ~= 11us; WMMA (GF(2) GEMM
// reformulation, 4.6e11 MACs) would be slower than this floor, so the
// CDNA5 paths used are the Tensor Data Mover (async perm->LDS stage,
// s_wait_tensorcnt) and wave32 ballot, not matrix ops.
// ---------------------------------------------------------------------------

#define KBITS 6144
#define ROWS  2048
#define ROWS_PER_BLOCK 8
#define THREADS (ROWS_PER_BLOCK * 32)
#define CHUNK 192      // bits handled per lane (KBITS / 32)
#define WORDS 6        // 32-bit words per lane (CHUNK / 32)
#define TPAD 33        // padded stride for transposed perm (conflict-free)

typedef unsigned int u32x4_t __attribute__((ext_vector_type(4)));
typedef int          i32x8_t __attribute__((ext_vector_type(8)));
typedef int          i32x4_t __attribute__((ext_vector_type(4)));
typedef float        f32x4_t __attribute__((ext_vector_type(4)));

__device__ __forceinline__ uint32_t ballot32(int pred) {
#if __has_builtin(__builtin_amdgcn_ballot_w32)
  return __builtin_amdgcn_ballot_w32(pred != 0);
#else
  return (uint32_t)__ballot(pred);
#endif
}

__global__ __launch_bounds__(THREADS) void turbo_encode_kernel(
    const float* __restrict__ bits, const int* __restrict__ perm,
    float* __restrict__ out) {
  __shared__ int permLin[KBITS];          // straight copy (TDM destination)
  __shared__ int permT[CHUNK * TPAD];     // transposed + padded (bank-friendly)

  const int tid  = threadIdx.x;
  const int lane = tid & 31;
  const int wave = tid >> 5;

  // ---- Stage perm into LDS with the Tensor Data Mover (async DMA) --------
#if __has_builtin(__builtin_amdgcn_tensor_load_to_lds) && \
    __has_builtin(__builtin_amdgcn_s_wait_tensorcnt)
  if (wave == 0) {
    // D# per CDNA5 ISA ch.8: group0 = {count/flags, lds_addr, global_addr, type}
    uint32_t lds_off = (uint32_t)(uintptr_t)(&permLin[0]); // flat LDS addr low 32b == LDS byte offset
    uint64_t ga = (uint64_t)(uintptr_t)perm;
    u32x4_t g0;
    g0[0] = 1u;                                         // count=1 (valid), no gather/iterate
    g0[1] = lds_off;                                    // lds_addr
    g0[2] = (uint32_t)(ga & 0xFFFFFFFFu);               // global_addr[31:0]
    g0[3] = (uint32_t)((ga >> 32) & 0x01FFFFFFu)        // global_addr[56:32]
          | (2u << 30);                                 // type = 2 ("image")
    i32x8_t g1;
    g1[0] = (int)(2u << 16);                            // workgroup_mask=0, data_size=2 (4B)
    g1[1] = (int)((KBITS & 0xFFFFu) << 16);             // tensor_dim0 lo16 = 6144
    g1[2] = (int)(1u << 16);                            // tensor_dim0 hi=0 | tensor_dim1 lo=1
    g1[3] = (int)((KBITS & 0xFFFFu) << 16);             // tensor_dim1 hi=0 | tile_dim0 = 6144
    g1[4] = 1;                                          // tile_dim1=1, tile_dim2=0
    g1[5] = KBITS;                                      // tensor_dim0_stride lo32
    g1[6] = (int)((KBITS & 0xFFFFu) << 16);             // stride0 hi=0 | stride1 lo16
    g1[7] = 0;                                          // stride1 hi
    i32x4_t gz4 = {0, 0, 0, 0};                         // groups 2/3 unused (<=2D)
    i32x8_t gz8 = {0, 0, 0, 0, 0, 0, 0, 0};             // extra operand (6-arg form), zero-filled
    __builtin_amdgcn_tensor_load_to_lds(g0, g1, gz4, gz4, gz8, 0);
    __builtin_amdgcn_s_wait_tensorcnt(0);
  }
#else
  for (int x = tid; x < KBITS; x += THREADS) permLin[x] = perm[x];
#endif
  __syncthreads();

  // ---- Transpose perm: permT[i*33 + c] = perm[c*192 + i] -----------------
  // (read LDS coalesced, write stride 33 -> both sides conflict-free)
  for (int x = tid; x < KBITS; x += THREADS) {
    int c = x / CHUNK;
    int i = x - c * CHUNK;
    permT[i * TPAD + c] = permLin[x];
  }
  __syncthreads();

  const int row = blockIdx.x * ROWS_PER_BLOCK + wave;   // one wave per row
  const float* __restrict__ urow = bits + (size_t)row * KBITS;
  const int base = lane * CHUNK;

  // ---- Pass A: pack natural-order bits (u) and interleaved bits (v) ------
  uint32_t uw[WORDS], vw[WORDS];
#pragma unroll
  for (int w = 0; w < WORDS; ++w) {
    uint32_t acc = 0;
    const f32x4_t* p4 = (const f32x4_t*)(urow + base + w * 32);
#pragma unroll
    for (int q = 0; q < 8; ++q) {
      f32x4_t f = p4[q];
      acc |= (uint32_t)(f.x > 0.5f) << (4 * q + 0);
      acc |= (uint32_t)(f.y > 0.5f) << (4 * q + 1);
      acc |= (uint32_t)(f.z > 0.5f) << (4 * q + 2);
      acc |= (uint32_t)(f.w > 0.5f) << (4 * q + 3);
    }
    uw[w] = acc;
  }
#pragma unroll
  for (int w = 0; w < WORDS; ++w) {
    uint32_t acc = 0;
#pragma unroll 8
    for (int b = 0; b < 32; ++b) {
      int i   = w * 32 + b;
      int idx = permT[i * TPAD + lane];      // conflict-free LDS read
      float f = urow[idx];                   // row window is L2/WGP$ resident
      acc |= (uint32_t)(f > 0.5f) << b;
    }
    vw[w] = acc;
  }

  // ---- Parity-class XOR totals + wave32 ballot exclusive prefix ----------
  const uint32_t M55 = 0x55555555u, MAA = 0xAAAAAAAAu;
  uint32_t tEu = 0, tOu = 0, tEv = 0, tOv = 0;
#pragma unroll
  for (int w = 0; w < WORDS; ++w) {
    tEu += __popc(uw[w] & M55);  tOu += __popc(uw[w] & MAA);
    tEv += __popc(vw[w] & M55);  tOv += __popc(vw[w] & MAA);
  }
  tEu &= 1; tOu &= 1; tEv &= 1; tOv &= 1;
  const uint32_t ltmask = (1u << lane) - 1u;
  uint32_t rEu = __popc(ballot32(tEu) & ltmask) & 1u;
  uint32_t rOu = __popc(ballot32(tOu) & ltmask) & 1u;
  uint32_t rEv = __popc(ballot32(tEv) & ltmask) & 1u;
  uint32_t rOv = __popc(ballot32(tOv) & ltmask) & 1u;

  // ---- Pass B: word-parallel prefix, expand to floats, NT stores ---------
  float* __restrict__ orow = out + (size_t)row * (3 * KBITS) + (size_t)base * 3;
#pragma unroll
  for (int w = 0; w < WORDS; ++w) {
    uint32_t u = uw[w], v = vw[w];
    uint32_t x = u << 1; x ^= x << 2; x ^= x << 4; x ^= x << 8; x ^= x << 16;
    uint32_t p1 = u ^ x ^ ((rOu ? M55 : 0u) ^ (rEu ? MAA : 0u));
    uint32_t y = v << 1; y ^= y << 2; y ^= y << 4; y ^= y << 8; y ^= y << 16;
    uint32_t p2 = v ^ y ^ ((rOv ? M55 : 0u) ^ (rEv ? MAA : 0u));
    rEu ^= __popc(u & M55) & 1u;  rOu ^= __popc(u & MAA) & 1u;
    rEv ^= __popc(v & M55) & 1u;  rOv ^= __popc(v & MAA) & 1u;

    float* ob = orow + w * 96;                 // 32 steps * 3 floats
#pragma unroll
    for (int b = 0; b < 32; b += 4) {
      f32x4_t f0, f1, f2;                      // 4 steps -> 12 floats, 16B aligned
      f0.x = (float)((u  >> (b + 0)) & 1u);
      f0.y = (float)((p1 >> (b + 0)) & 1u);
      f0.z = (float)((p2 >> (b + 0)) & 1u);
      f0.w = (float)((u  >> (b + 1)) & 1u);
      f1.x = (float)((p1 >> (b + 1)) & 1u);
      f1.y = (float)((p2 >> (b + 1)) & 1u);
      f1.z = (float)((u  >> (b + 2)) & 1u);
      f1.w = (float)((p1 >> (b + 2)) & 1u);
      f2.x = (float)((p2 >> (b + 2)) & 1u);
      f2.y = (float)((u  >> (b + 3)) & 1u);
      f2.z = (float)((p1 >> (b + 3)) & 1u);
      f2.w = (float)((p2 >> (b + 3)) & 1u);
      f32x4_t* dst = (f32x4_t*)(ob + b * 3);
      __builtin_nontemporal_store(f0, dst + 0);
      __builtin_nontemporal_store(f1, dst + 1);
      __builtin_nontemporal_store(f2, dst + 2);
    }
  }
}

extern "C" void kernel_launch(void* const* d_in, const int* in_sizes, int n_in,
                              void* d_out, int out_size, void* d_ws,
                              size_t ws_size, hipStream_t stream) {
  (void)in_sizes; (void)n_in; (void)d_ws; (void)ws_size; (void)out_size;
  const float* bits = (const float*)d_in[0];
  const int*   perm = (const int*)d_in[1];
  float*       out  = (float*)d_out;
  dim3 grid(ROWS / ROWS_PER_BLOCK);   // 256 workgroups x 8 waves = 2048 rows
  turbo_encode_kernel<<<grid, THREADS, 0, stream>>>(bits, perm, out);
}